// YoloLoss_31980326486022
// MI455X (gfx1250) — compile-verified
//
#include <hip/hip_runtime.h>
#include <hip/hip_bf16.h>

// ---------------------------------------------------------------------------
// YOLO loss reduction for MI455X (gfx1250, wave32).
// Memory-bound streaming reduction: ~174MB in, 5 floats out (~7.5us at 23.3TB/s).
// Cross-lane sums use V_WMMA_F32_16X16X4_F32 with B=ones (layout-independent).
// ---------------------------------------------------------------------------

typedef __attribute__((ext_vector_type(2))) float v2f;
typedef __attribute__((ext_vector_type(8))) float v8f;

#define S_DIM   14
#define NCLS    20
#define CH      30           // B*5 + NCLS
#define BLOCK   256
#define GRID    1024
#define L_COORD 5.0f
#define L_NOOBJ 0.5f

// Full-wave (32 lane) sum via one WMMA.
//   A[m][0] = v(lane m), A[m][2] = v(lane m+16)  (documented 16x4 f32 A layout)
//   B = all ones  -> D[m][n] = v(m) + v(m+16) for every n.
//   D layout: vgpr r, lanes 0..15 -> row r ; lanes 16..31 -> row r+8.
//   sum_r d[r] gives half-wave sums; fold halves with shfl_xor(16).
// Requires EXEC == all ones in the executing wave.
__device__ __forceinline__ float wave_sum_wmma(float v) {
    v2f a; a.x = v;    a.y = 0.0f;
    v2f b; b.x = 1.0f; b.y = 1.0f;
    v8f c = {};
    c = __builtin_amdgcn_wmma_f32_16x16x4_f32(
            /*neg_a=*/false, a, /*neg_b=*/false, b,
            /*c_mod=*/(short)0, c, /*reuse_a=*/false, /*reuse_b=*/false);
    float s = ((c[0] + c[1]) + (c[2] + c[3])) + ((c[4] + c[5]) + (c[6] + c[7]));
    s += __shfl_xor(s, 16, 32);
    return s;   // full 32-lane sum, broadcast to all lanes
}

__global__ __launch_bounds__(BLOCK)
void yolo_partials(const float* __restrict__ pred,
                   const float* __restrict__ tbox,
                   const float* __restrict__ tcls,
                   const unsigned char* __restrict__ objmap,
                   float* __restrict__ ws,
                   int ncells) {
    float acc_cls = 0.0f, acc_noobj = 0.0f, acc_obj = 0.0f, acc_reg = 0.0f;

    for (int idx = blockIdx.x * BLOCK + threadIdx.x; idx < ncells;
         idx += GRID * BLOCK) {
        const float* p  = pred + idx * CH;
        const float* tb = tbox + idx * 4;
        const float* tc = tcls + idx * NCLS;
        const float  m  = objmap[idx] ? 1.0f : 0.0f;

        // ---- class loss: m * sum_c (pred_cls - target_cls)^2 ----
        float cl = 0.0f;
        #pragma unroll
        for (int c = 0; c < NCLS; ++c) {
            float d = p[10 + c] - tc[c];
            cl = fmaf(d, d, cl);
        }
        acc_cls += m * cl;

        // ---- targets ----
        const float t0 = tb[0], t1 = tb[1], t2 = tb[2], t3 = tb[3];
        const float area_t = (t2 - t0) * (t3 - t1);

        // ---- box 0 ----
        const float x0 = p[0], y0 = p[1], w0 = p[2], h0 = p[3], c0 = p[4];
        // ---- box 1 ----
        const float x1 = p[5], y1 = p[6], w1 = p[7], h1 = p[8], c1 = p[9];

        // no-object loss: (1-m) * (conf0^2 + conf1^2)
        acc_noobj += (1.0f - m) * (c0 * c0 + c1 * c1);

        const float invS = 1.0f / (float)S_DIM;

        // IOU box 0
        float cx = x0 * invS, cy = y0 * invS;
        float ltx = fmaxf(cx - 0.5f * w0, t0);
        float lty = fmaxf(cy - 0.5f * h0, t1);
        float rbx = fminf(cx + 0.5f * w0, t2);
        float rby = fminf(cy + 0.5f * h0, t3);
        float iw  = fmaxf(rbx - ltx, 0.0f);
        float ih  = fmaxf(rby - lty, 0.0f);
        float inter = iw * ih;
        float iou0  = inter / (w0 * h0 + area_t - inter);

        // IOU box 1
        cx = x1 * invS; cy = y1 * invS;
        ltx = fmaxf(cx - 0.5f * w1, t0);
        lty = fmaxf(cy - 0.5f * h1, t1);
        rbx = fminf(cx + 0.5f * w1, t2);
        rby = fminf(cy + 0.5f * h1, t3);
        iw  = fmaxf(rbx - ltx, 0.0f);
        ih  = fmaxf(rby - lty, 0.0f);
        inter = iw * ih;
        float iou1 = inter / (w1 * h1 + area_t - inter);

        // argmax (first index wins ties, matching jnp.argmax)
        const bool b1 = iou1 > iou0;
        const float bx = b1 ? x1 : x0;
        const float by = b1 ? y1 : y0;
        const float bw = b1 ? w1 : w0;
        const float bh = b1 ? h1 : h0;
        const float bc = b1 ? c1 : c0;

        // containing-object loss: m * (conf - 1)^2
        const float dc = bc - 1.0f;
        acc_obj += m * dc * dc;

        // regression loss: m * (center + dim)
        const float dx = bx - t0;
        const float dy = by - t1;
        const float dw = __builtin_sqrtf(bw) - __builtin_sqrtf(t2);
        const float dh = __builtin_sqrtf(bh) - __builtin_sqrtf(t3);
        acc_reg += m * (dx * dx + dy * dy + dw * dw + dh * dh);
    }

    // ---- block reduction: LDS stage, then WMMA wave reduction by wave 0 ----
    __shared__ float sh[4][BLOCK];
    const int tid = threadIdx.x;
    sh[0][tid] = acc_cls;
    sh[1][tid] = acc_noobj;
    sh[2][tid] = acc_obj;
    sh[3][tid] = acc_reg;
    __syncthreads();

    if (tid < 32) {               // wave-uniform branch: wave 0 has EXEC = ~0
        float p0 = 0.0f, p1 = 0.0f, p2 = 0.0f, p3 = 0.0f;
        #pragma unroll
        for (int w = 0; w < BLOCK / 32; ++w) {
            p0 += sh[0][tid + 32 * w];
            p1 += sh[1][tid + 32 * w];
            p2 += sh[2][tid + 32 * w];
            p3 += sh[3][tid + 32 * w];
        }
        p0 = wave_sum_wmma(p0);
        p1 = wave_sum_wmma(p1);
        p2 = wave_sum_wmma(p2);
        p3 = wave_sum_wmma(p3);
        if (tid == 0) {
            float* dst = ws + blockIdx.x * 4;
            dst[0] = p0; dst[1] = p1; dst[2] = p2; dst[3] = p3;
        }
    }
}

__global__ __launch_bounds__(BLOCK)
void yolo_finalize(const float* __restrict__ ws,
                   float* __restrict__ out,
                   int nblocks, float inv_n) {
    const int tid = threadIdx.x;
    float a0 = 0.0f, a1 = 0.0f, a2 = 0.0f, a3 = 0.0f;
    for (int b = tid; b < nblocks; b += BLOCK) {
        const float* src = ws + b * 4;
        a0 += src[0]; a1 += src[1]; a2 += src[2]; a3 += src[3];
    }

    __shared__ float sh[4][BLOCK];
    sh[0][tid] = a0; sh[1][tid] = a1; sh[2][tid] = a2; sh[3][tid] = a3;
    __syncthreads();

    if (tid < 32) {
        float p0 = 0.0f, p1 = 0.0f, p2 = 0.0f, p3 = 0.0f;
        #pragma unroll
        for (int w = 0; w < BLOCK / 32; ++w) {
            p0 += sh[0][tid + 32 * w];
            p1 += sh[1][tid + 32 * w];
            p2 += sh[2][tid + 32 * w];
            p3 += sh[3][tid + 32 * w];
        }
        p0 = wave_sum_wmma(p0);   // cls
        p1 = wave_sum_wmma(p1);   // noobj
        p2 = wave_sum_wmma(p2);   // obj
        p3 = wave_sum_wmma(p3);   // reg
        if (tid == 0) {
            const float cls   = p0 * inv_n;
            const float noobj = p1 * inv_n;
            const float obj   = p2 * inv_n;
            const float reg   = p3 * inv_n;
            out[0] = cls + L_NOOBJ * noobj + L_COORD * reg + obj;  // total
            out[1] = reg;
            out[2] = obj;
            out[3] = noobj;
            out[4] = cls;
        }
    }
}

extern "C" void kernel_launch(void* const* d_in, const int* in_sizes, int n_in,
                              void* d_out, int out_size, void* d_ws, size_t ws_size,
                              hipStream_t stream) {
    const float* pred = (const float*)d_in[0];           // (N,14,14,30) f32
    const float* tbox = (const float*)d_in[1];           // (N,14,14,4)  f32
    const float* tcls = (const float*)d_in[2];           // (N,14,14,20) f32
    const unsigned char* objmap = (const unsigned char*)d_in[3]; // (N,14,14) bool (1B)
    float* out = (float*)d_out;                          // 5 f32
    float* ws  = (float*)d_ws;                           // GRID*4 f32 partials

    const int ncells = in_sizes[3];                      // N*14*14
    const float inv_n = (float)(S_DIM * S_DIM) / (float)ncells; // 1/N

    yolo_partials<<<GRID, BLOCK, 0, stream>>>(pred, tbox, tcls, objmap, ws, ncells);
    yolo_finalize<<<1, BLOCK, 0, stream>>>(ws, out, GRID, inv_n);
}